// Fusionv2_11699490914865
// MI455X (gfx1250) — compile-verified
//
#include <hip/hip_runtime.h>
#include <math.h>

#define BATCH 8
#define CH 64
#define WDIM 256
#define WH 65536
#define TP 32            // flat positions per block in proj/out kernels
#define SLABS 64         // K-slabs per batch in QK kernel
#define SK (WH / SLABS)  // 1024
#define CKK 64           // K-chunk staged in LDS per iteration
#define NCHUNK (SK / CKK)  // 16
#define QPAD 4           // 16B-aligned LDS rows (68 floats), conflict-free

typedef __attribute__((ext_vector_type(2))) float v2f;
typedef __attribute__((ext_vector_type(8))) float v8f;

__device__ __forceinline__ v8f wmma_f32_4(v2f a, v2f b, v8f c) {
  // D(16x16,f32) = A(16x4,f32) * B(4x16,f32) + C
  return __builtin_amdgcn_wmma_f32_16x16x4_f32(false, a, false, b, (short)0, c,
                                               false, false);
}

#if defined(__gfx1250__) &&                                         \
    __has_builtin(__builtin_amdgcn_global_load_async_to_lds_b128) && \
    __has_builtin(__builtin_amdgcn_s_wait_asynccnt)
#define USE_ASYNC_LDS 1
typedef int gv4i __attribute__((vector_size(16)));
typedef __attribute__((address_space(3))) gv4i lds_v4i;
__device__ __forceinline__ void async_copy_f4(const float* g, float* l) {
  __builtin_amdgcn_global_load_async_to_lds_b128((gv4i*)g, (lds_v4i*)l, 0, 0);
}
#endif

// ---------------------------------------------------------------------------
// Phase 1: fused depthwise 3x3 + pointwise 1x1 (WMMA) for q_x, k_x, q_m
// grid: (WH/TP, BATCH), block 256 (8 waves)
// ---------------------------------------------------------------------------
__global__ __launch_bounds__(256)
void proj_qkm_kernel(const float* __restrict__ x, const float* __restrict__ mask,
                     const float* __restrict__ dw_q, const float* __restrict__ db_q,
                     const float* __restrict__ pw_q, const float* __restrict__ pb_q,
                     const float* __restrict__ dw_k, const float* __restrict__ db_k,
                     const float* __restrict__ pw_k, const float* __restrict__ pb_k,
                     float* __restrict__ qx, float* __restrict__ kx,
                     float* __restrict__ qm) {
  __shared__ float tq[CH][TP + 1];
  __shared__ float tk[CH][TP + 1];
  __shared__ float tm[CH][TP + 1];

  const int tid = threadIdx.x;
  const int b = blockIdx.y;
  const int p0 = blockIdx.x * TP;  // 32 flat positions, all in one image row
  const int row = p0 >> 8;
  const int col0 = p0 & 255;
  const size_t base_b = (size_t)b * CH * WH;

  // depthwise 3x3 (SAME, zero pad) for q(x), k(x), q(mask)
  #pragma unroll
  for (int i = 0; i < 8; ++i) {
    int idx = tid + 256 * i;  // 0..2047 = 64ch x 32pos
    int ch = idx >> 5;
    int pp = idx & 31;
    int col = col0 + pp;
    const float* xin = x + base_b + (size_t)ch * WH;
    const float* min_ = mask + base_b + (size_t)ch * WH;
    float aq = 0.f, ak = 0.f, am = 0.f;
    #pragma unroll
    for (int kr = 0; kr < 3; ++kr) {
      int rr = row + kr - 1;
      bool rok = (rr >= 0) && (rr < WDIM);
      #pragma unroll
      for (int kc = 0; kc < 3; ++kc) {
        int cc = col + kc - 1;
        bool ok = rok && (cc >= 0) && (cc < WDIM);
        float xv = ok ? xin[rr * WDIM + cc] : 0.f;
        float mv = ok ? min_[rr * WDIM + cc] : 0.f;
        float wq = dw_q[ch * 9 + kr * 3 + kc];
        float wk = dw_k[ch * 9 + kr * 3 + kc];
        aq += xv * wq;
        ak += xv * wk;
        am += mv * wq;
      }
    }
    tq[ch][pp] = aq + db_q[ch];
    tk[ch][pp] = ak + db_k[ch];
    tm[ch][pp] = am + db_q[ch];
  }
  __syncthreads();

  // pointwise: OUT(64 x 32) = PW(64x64) @ T(64x32) + bias
  const int wv = tid >> 5;
  const int lane = tid & 31;
  const int mt = wv >> 1;  // 4 M-tiles of 16 out-channels
  const int nt = wv & 1;   // 2 N-tiles of 16 positions
  const int lm = lane & 15;
  const int lh = lane >> 4;

  for (int pj = 0; pj < 3; ++pj) {
    const float* pw = (pj == 1) ? pw_k : pw_q;
    const float* pb = (pj == 1) ? pb_k : pb_q;
    const float(*Ts)[TP + 1] = (pj == 0) ? tq : (pj == 1) ? tk : tm;
    float* outp = (pj == 0) ? qx : (pj == 1) ? kx : qm;

    v8f acc;
    #pragma unroll
    for (int r = 0; r < 8; ++r) acc[r] = pb[mt * 16 + r + 8 * lh];
    #pragma unroll
    for (int k4 = 0; k4 < 16; ++k4) {
      int kb = k4 * 4;
      v2f a, bf;
      a.x = pw[(mt * 16 + lm) * 64 + kb + 2 * lh];
      a.y = pw[(mt * 16 + lm) * 64 + kb + 2 * lh + 1];
      bf.x = Ts[kb + 2 * lh][nt * 16 + lm];
      bf.y = Ts[kb + 2 * lh + 1][nt * 16 + lm];
      acc = wmma_f32_4(a, bf, acc);
    }
    #pragma unroll
    for (int r = 0; r < 8; ++r) {
      int ch = mt * 16 + r + 8 * lh;
      outp[base_b + (size_t)ch * WH + p0 + nt * 16 + lm] = acc[r];
    }
  }
}

// ---------------------------------------------------------------------------
// Phase 2: qk1 = q_x @ K', qk2 = q_m @ K' with K'[p,j] = kflat[p*64+j].
// grid: (SLABS, BATCH), block 256 (8 waves). Deterministic per-slab partials.
// Double-buffered LDS; async global->LDS (B128) when toolchain exposes it.
// ---------------------------------------------------------------------------
__global__ __launch_bounds__(256)
void qk_kernel(const float* __restrict__ qx, const float* __restrict__ qm,
               const float* __restrict__ kx,
               float* __restrict__ qkp1, float* __restrict__ qkp2) {
  __shared__ float Qs1[2][CH][CKK + QPAD];
  __shared__ float Qs2[2][CH][CKK + QPAD];
  __shared__ float Bs[2][CKK][CH + QPAD];

  const int tid = threadIdx.x;
  const int b = blockIdx.y;
  const int k0 = blockIdx.x * SK;
  const size_t base_b = (size_t)b * CH * WH;

  const int wv = tid >> 5;
  const int lane = tid & 31;
  const int lm = lane & 15;
  const int lh = lane >> 4;
  const int mtx = wv >> 2;  // 0 -> qk1, 1 -> qk2
  const int mt = wv & 3;    // M-tile

  v8f acc[4];
  #pragma unroll
  for (int n = 0; n < 4; ++n)
    #pragma unroll
    for (int r = 0; r < 8; ++r) acc[n][r] = 0.0f;

#if defined(USE_ASYNC_LDS)
  // ---- async double-buffered pipeline: 12 B128 ops per chunk per wave ----
  auto issue_chunk = [&](int ci, int buf) {
    const int kc0 = k0 + ci * CKK;
    #pragma unroll
    for (int l = 0; l < 4; ++l) {
      int fi = tid + 256 * l;      // float4 index 0..1023
      int rr = fi >> 4;            // row 0..63
      int c4 = (fi & 15) * 4;      // col (float), multiple of 4 -> 16B aligned
      async_copy_f4(qx + base_b + (size_t)rr * WH + kc0 + c4, &Qs1[buf][rr][c4]);
      async_copy_f4(qm + base_b + (size_t)rr * WH + kc0 + c4, &Qs2[buf][rr][c4]);
      async_copy_f4(kx + base_b + (size_t)kc0 * 64 + (size_t)fi * 4,
                    &Bs[buf][rr][c4]);
    }
  };
  issue_chunk(0, 0);
  for (int ci = 0; ci < NCHUNK; ++ci) {
    const int cur = ci & 1;
    if (ci + 1 < NCHUNK) {
      issue_chunk(ci + 1, cur ^ 1);
      __builtin_amdgcn_s_wait_asynccnt(12);  // oldest chunk landed
    } else {
      __builtin_amdgcn_s_wait_asynccnt(0);
    }
    __syncthreads();
    const float(*Qs)[CKK + QPAD] = mtx ? Qs2[cur] : Qs1[cur];
    const float(*Bb)[CH + QPAD] = Bs[cur];
    #pragma unroll 4
    for (int k4 = 0; k4 < 16; ++k4) {
      int kb = k4 * 4;
      v2f a;
      a.x = Qs[mt * 16 + lm][kb + 2 * lh];
      a.y = Qs[mt * 16 + lm][kb + 2 * lh + 1];
      #pragma unroll
      for (int n = 0; n < 4; ++n) {
        v2f bf;
        bf.x = Bb[kb + 2 * lh][n * 16 + lm];
        bf.y = Bb[kb + 2 * lh + 1][n * 16 + lm];
        acc[n] = wmma_f32_4(a, bf, acc[n]);
      }
    }
    __syncthreads();  // protect buffer reuse by next issue
  }
#else
  // ---- fallback: synchronous staging through VGPRs ----
  for (int ci = 0; ci < NCHUNK; ++ci) {
    const int cur = ci & 1;
    const int kc0 = k0 + ci * CKK;
    #pragma unroll
    for (int l = 0; l < 16; ++l) {
      int e = tid + 256 * l;  // 0..4095
      int rr = e >> 6;
      int cc = e & 63;
      Qs1[cur][rr][cc] = qx[base_b + (size_t)rr * WH + kc0 + cc];
      Qs2[cur][rr][cc] = qm[base_b + (size_t)rr * WH + kc0 + cc];
      Bs[cur][rr][cc] = kx[base_b + (size_t)kc0 * 64 + e];
    }
    __syncthreads();
    const float(*Qs)[CKK + QPAD] = mtx ? Qs2[cur] : Qs1[cur];
    const float(*Bb)[CH + QPAD] = Bs[cur];
    #pragma unroll 4
    for (int k4 = 0; k4 < 16; ++k4) {
      int kb = k4 * 4;
      v2f a;
      a.x = Qs[mt * 16 + lm][kb + 2 * lh];
      a.y = Qs[mt * 16 + lm][kb + 2 * lh + 1];
      #pragma unroll
      for (int n = 0; n < 4; ++n) {
        v2f bf;
        bf.x = Bb[kb + 2 * lh][n * 16 + lm];
        bf.y = Bb[kb + 2 * lh + 1][n * 16 + lm];
        acc[n] = wmma_f32_4(a, bf, acc[n]);
      }
    }
    __syncthreads();
  }
#endif

  // deterministic per-slab partial store (reduced in softmax kernel)
  float* dst = (mtx ? qkp2 : qkp1) +
               ((size_t)blockIdx.x * BATCH + b) * 4096;
  #pragma unroll
  for (int n = 0; n < 4; ++n)
    #pragma unroll
    for (int r = 0; r < 8; ++r) {
      int rowo = mt * 16 + r + 8 * lh;
      int colo = n * 16 + lm;
      dst[rowo * 64 + colo] = acc[n][r];
    }
}

// ---------------------------------------------------------------------------
// Phase 3: reduce slabs, then A = softmax(qk1/256 over flat 4096)
//                                + softmax(qk2/256).  grid: BATCH
// ---------------------------------------------------------------------------
__global__ __launch_bounds__(256)
void softmax_add_kernel(const float* __restrict__ qkp1,
                        const float* __restrict__ qkp2, float* __restrict__ A) {
  __shared__ float red[256];
  const int b = blockIdx.x;
  const int tid = threadIdx.x;
  const float scale = 1.0f / 256.0f;  // 1/sqrt(65536)
  float v1[16], v2[16];
  #pragma unroll
  for (int i = 0; i < 16; ++i) {
    int idx = tid + 256 * i;
    float s1 = 0.f, s2 = 0.f;
    for (int s = 0; s < SLABS; ++s) {  // fixed order: deterministic
      size_t o = ((size_t)s * BATCH + b) * 4096 + idx;
      s1 += qkp1[o];
      s2 += qkp2[o];
    }
    v1[i] = s1 * scale;
    v2[i] = s2 * scale;
  }
  float m1 = -3.4e38f, m2 = -3.4e38f;
  #pragma unroll
  for (int i = 0; i < 16; ++i) {
    m1 = fmaxf(m1, v1[i]);
    m2 = fmaxf(m2, v2[i]);
  }
  red[tid] = m1; __syncthreads();
  for (int s = 128; s; s >>= 1) { if (tid < s) red[tid] = fmaxf(red[tid], red[tid + s]); __syncthreads(); }
  m1 = red[0]; __syncthreads();
  red[tid] = m2; __syncthreads();
  for (int s = 128; s; s >>= 1) { if (tid < s) red[tid] = fmaxf(red[tid], red[tid + s]); __syncthreads(); }
  m2 = red[0]; __syncthreads();

  float s1 = 0.f, s2 = 0.f;
  #pragma unroll
  for (int i = 0; i < 16; ++i) {
    v1[i] = __expf(v1[i] - m1);
    v2[i] = __expf(v2[i] - m2);
    s1 += v1[i];
    s2 += v2[i];
  }
  red[tid] = s1; __syncthreads();
  for (int s = 128; s; s >>= 1) { if (tid < s) red[tid] += red[tid + s]; __syncthreads(); }
  s1 = red[0]; __syncthreads();
  red[tid] = s2; __syncthreads();
  for (int s = 128; s; s >>= 1) { if (tid < s) red[tid] += red[tid + s]; __syncthreads(); }
  s2 = red[0];

  float r1 = 1.0f / s1, r2 = 1.0f / s2;
  float* pa = A + (size_t)b * 4096;
  #pragma unroll
  for (int i = 0; i < 16; ++i) pa[tid + 256 * i] = v1[i] * r1 + v2[i] * r2;
}

// ---------------------------------------------------------------------------
// Phase 4: v_s = dsc(x+avg; dw_v, 2*db_v, pw_v, 2*pb_v); out = A @ v_s + x
// grid: (WH/TP, BATCH), block 256
// ---------------------------------------------------------------------------
__global__ __launch_bounds__(256)
void out_kernel(const float* __restrict__ x, const float* __restrict__ avg,
                const float* __restrict__ dw_v, const float* __restrict__ db_v,
                const float* __restrict__ pw_v, const float* __restrict__ pb_v,
                const float* __restrict__ Amat, float* __restrict__ out) {
  __shared__ float tv[CH][TP + 1];
  __shared__ float vs[CH][TP + 1];

  const int tid = threadIdx.x;
  const int b = blockIdx.y;
  const int p0 = blockIdx.x * TP;
  const int row = p0 >> 8;
  const int col0 = p0 & 255;
  const size_t base_b = (size_t)b * CH * WH;

  // depthwise on (x + avg_ct) with doubled bias
  #pragma unroll
  for (int i = 0; i < 8; ++i) {
    int idx = tid + 256 * i;
    int ch = idx >> 5;
    int pp = idx & 31;
    int col = col0 + pp;
    const float* xin = x + base_b + (size_t)ch * WH;
    const float* ain = avg + base_b + (size_t)ch * WH;
    float av = 0.f;
    #pragma unroll
    for (int kr = 0; kr < 3; ++kr) {
      int rr = row + kr - 1;
      bool rok = (rr >= 0) && (rr < WDIM);
      #pragma unroll
      for (int kc = 0; kc < 3; ++kc) {
        int cc = col + kc - 1;
        bool ok = rok && (cc >= 0) && (cc < WDIM);
        float sv = ok ? (xin[rr * WDIM + cc] + ain[rr * WDIM + cc]) : 0.f;
        av += sv * dw_v[ch * 9 + kr * 3 + kc];
      }
    }
    tv[ch][pp] = av + 2.0f * db_v[ch];
  }
  __syncthreads();

  const int wv = tid >> 5;
  const int lane = tid & 31;
  const int mt = wv >> 1;
  const int nt = wv & 1;
  const int lm = lane & 15;
  const int lh = lane >> 4;

  // pointwise: v_s tile = PW_v @ tv + 2*pb_v  -> vs in LDS
  {
    v8f acc;
    #pragma unroll
    for (int r = 0; r < 8; ++r) acc[r] = 2.0f * pb_v[mt * 16 + r + 8 * lh];
    #pragma unroll
    for (int k4 = 0; k4 < 16; ++k4) {
      int kb = k4 * 4;
      v2f a, bf;
      a.x = pw_v[(mt * 16 + lm) * 64 + kb + 2 * lh];
      a.y = pw_v[(mt * 16 + lm) * 64 + kb + 2 * lh + 1];
      bf.x = tv[kb + 2 * lh][nt * 16 + lm];
      bf.y = tv[kb + 2 * lh + 1][nt * 16 + lm];
      acc = wmma_f32_4(a, bf, acc);
    }
    #pragma unroll
    for (int r = 0; r < 8; ++r)
      vs[mt * 16 + r + 8 * lh][nt * 16 + lm] = acc[r];
  }
  __syncthreads();

  // second GEMM: out tile = A(64x64) @ vs(64x32) + x residual
  {
    const float* Ab = Amat + (size_t)b * 4096;
    v8f acc;
    #pragma unroll
    for (int r = 0; r < 8; ++r) acc[r] = 0.0f;
    #pragma unroll
    for (int k4 = 0; k4 < 16; ++k4) {
      int kb = k4 * 4;
      v2f a, bf;
      a.x = Ab[(mt * 16 + lm) * 64 + kb + 2 * lh];
      a.y = Ab[(mt * 16 + lm) * 64 + kb + 2 * lh + 1];
      bf.x = vs[kb + 2 * lh][nt * 16 + lm];
      bf.y = vs[kb + 2 * lh + 1][nt * 16 + lm];
      acc = wmma_f32_4(a, bf, acc);
    }
    #pragma unroll
    for (int r = 0; r < 8; ++r) {
      int ch = mt * 16 + r + 8 * lh;
      size_t o = base_b + (size_t)ch * WH + p0 + nt * 16 + lm;
      out[o] = acc[r] + x[o];
    }
  }
}

// ---------------------------------------------------------------------------
extern "C" void kernel_launch(void* const* d_in, const int* in_sizes, int n_in,
                              void* d_out, int out_size, void* d_ws,
                              size_t ws_size, hipStream_t stream) {
  (void)in_sizes; (void)n_in; (void)out_size; (void)ws_size;
  const float* x    = (const float*)d_in[0];
  const float* mask = (const float*)d_in[1];
  const float* avg  = (const float*)d_in[2];
  const float* dw_q = (const float*)d_in[3];
  const float* db_q = (const float*)d_in[4];
  const float* pw_q = (const float*)d_in[5];
  const float* pb_q = (const float*)d_in[6];
  const float* dw_k = (const float*)d_in[7];
  const float* db_k = (const float*)d_in[8];
  const float* pw_k = (const float*)d_in[9];
  const float* pb_k = (const float*)d_in[10];
  const float* dw_v = (const float*)d_in[11];
  const float* db_v = (const float*)d_in[12];
  const float* pw_v = (const float*)d_in[13];
  const float* pb_v = (const float*)d_in[14];
  float* out = (float*)d_out;
  float* ws = (float*)d_ws;

  const size_t N = (size_t)BATCH * CH * WH;  // 33,554,432 floats per tensor
  const size_t PART = (size_t)SLABS * BATCH * 4096;  // 2,097,152 floats
  float* qxp  = ws;
  float* kxp  = ws + N;
  float* qmp  = ws + 2 * N;
  float* qkp1 = ws + 3 * N;
  float* qkp2 = qkp1 + PART;
  float* Am   = qkp2 + PART;

  hipLaunchKernelGGL(proj_qkm_kernel, dim3(WH / TP, BATCH), dim3(256), 0,
                     stream, x, mask, dw_q, db_q, pw_q, pb_q, dw_k, db_k, pw_k,
                     pb_k, qxp, kxp, qmp);
  hipLaunchKernelGGL(qk_kernel, dim3(SLABS, BATCH), dim3(256), 0, stream, qxp,
                     qmp, kxp, qkp1, qkp2);
  hipLaunchKernelGGL(softmax_add_kernel, dim3(BATCH), dim3(256), 0, stream,
                     qkp1, qkp2, Am);
  hipLaunchKernelGGL(out_kernel, dim3(WH / TP, BATCH), dim3(256), 0, stream, x,
                     avg, dw_v, db_v, pw_v, pb_v, Am, out);
}